// K_Means_80401787781311
// MI455X (gfx1250) — compile-verified
//
#include <hip/hip_runtime.h>

// K-Means broadcast-subtract: out[k][n][d] = x[n][d] - c[k][d]
// N=65536 points, K=32 centroids, D=64 -> out = 512 MiB, pure streaming-store bound.

typedef float v4f __attribute__((ext_vector_type(4)));
typedef int   v4i __attribute__((ext_vector_type(4)));

#define NPTS 65536
#define KC   32
#define DW   64
#define D4   (DW / 4)          // 16 float4 chunks per row
#define PTS_PER_BLOCK 16
#define TPB  256               // 16 points x 16 chunks

__global__ __launch_bounds__(TPB) void kmeans_diff_kernel(
    const float* __restrict__ x,
    const float* __restrict__ cen,
    float* __restrict__ out)
{
    // 32 x 64 floats = 8 KB of centroids staged once per block
    __shared__ v4f cs[KC * D4];

    const int tid = threadIdx.x;
    const v4f* cg = (const v4f*)cen;

#if defined(__gfx1250__) && __has_builtin(__builtin_amdgcn_global_load_async_to_lds_b128)
    // CDNA5 async global->LDS path: each of the 256 lanes DMA-copies two
    // float4s global->LDS without touching VGPR data, tracked by ASYNCcnt.
    {
        __builtin_amdgcn_global_load_async_to_lds_b128(
            (v4i*)(cg + tid),        (v4i*)(cs + tid),        0, 0);
        __builtin_amdgcn_global_load_async_to_lds_b128(
            (v4i*)(cg + tid + TPB),  (v4i*)(cs + tid + TPB),  0, 0);
  #if __has_builtin(__builtin_amdgcn_s_wait_asynccnt)
        __builtin_amdgcn_s_wait_asynccnt(0);
  #else
        asm volatile("s_wait_asynccnt 0" ::: "memory");
  #endif
    }
#else
    // Fallback: plain cooperative copy (2 float4 per thread covers 512 float4)
    cs[tid]       = cg[tid];
    cs[tid + TPB] = cg[tid + TPB];
#endif
    __syncthreads();

    const int d4 = tid & (D4 - 1);            // which float4 chunk of D
    const int n  = blockIdx.x * PTS_PER_BLOCK + (tid >> 4);

    // x is read exactly once globally -> non-temporal 128-bit load
    const v4f xv = __builtin_nontemporal_load((const v4f*)x + (size_t)n * D4 + d4);

    // Output base for k=0; stride between k-slices is N*D floats (16 MiB)
    v4f* o = (v4f*)out + (size_t)n * D4 + d4;

#pragma unroll
    for (int k = 0; k < KC; ++k) {
        v4f cv = cs[k * D4 + d4];             // ds_load_b128, bank-conflict free
        v4f r  = xv - cv;
        // 512 MiB write-once stream: non-temporal 128-bit store (th:TH_STORE_NT)
        __builtin_nontemporal_store(r, o);
        o += (size_t)NPTS * D4;
    }
}

extern "C" void kernel_launch(void* const* d_in, const int* in_sizes, int n_in,
                              void* d_out, int out_size, void* d_ws, size_t ws_size,
                              hipStream_t stream) {
    (void)in_sizes; (void)n_in; (void)out_size; (void)d_ws; (void)ws_size;
    const float* x = (const float*)d_in[0];       // [N, D] float32
    const float* c = (const float*)d_in[1];       // [K, D] float32
    float* out     = (float*)d_out;               // [K, N, D] float32

    dim3 grid(NPTS / PTS_PER_BLOCK);              // 4096 blocks
    dim3 block(TPB);                              // 256 threads = 8 waves (wave32)
    kmeans_diff_kernel<<<grid, block, 0, stream>>>(x, c, out);
}